// GATLayer_81011673137389
// MI455X (gfx1250) — compile-verified
//
#include <hip/hip_runtime.h>
#include <hip/hip_bf16.h>

#define N_NODES  100000
#define N_FEAT   256
#define N_HIDDEN 64
#define N_EDGES  1600000
#define NEG_SLOPE 0.05f

typedef __attribute__((ext_vector_type(2))) float v2f;
typedef __attribute__((ext_vector_type(8))) float v8f;

// ---------------------------------------------------------------------------
// K1: Wh = x @ W_fc   via V_WMMA_F32_16X16X4_F32
// One wave computes a 16-row x 64-col tile (4 accumulators of 16x16).
// A fragment (16x4 f32): lane L (lo=L&15, hi=L>>4) holds x[row0+lo, k0+2*hi+{0,1}]
// B fragment (4x16 f32): lane L holds W[(k0+2*hi+{0,1})*64 + n], n = tile*16+lo
// C/D: VGPR r, lanes0-15 -> (M=r, N=lo); lanes16-31 -> (M=r+8, N=lo)
// ---------------------------------------------------------------------------
__global__ __launch_bounds__(256) void gemm_wh(const float* __restrict__ x,
                                               const float* __restrict__ W,
                                               float* __restrict__ Wh) {
    const int wave = threadIdx.x >> 5;
    const int lane = threadIdx.x & 31;
    const int tile = blockIdx.x * 8 + wave;          // 16-row tile index
    if (tile >= (N_NODES / 16)) return;              // wave-uniform exit (EXEC stays all-1s)

    const int row0 = tile * 16;
    const int lo = lane & 15;
    const int hi = lane >> 4;

    v8f acc0 = {}, acc1 = {}, acc2 = {}, acc3 = {};

    const float* xrow = x + (size_t)(row0 + lo) * N_FEAT + 2 * hi;

    for (int k0 = 0; k0 < N_FEAT; k0 += 4) {
        v2f a;
        const float2 av = *(const float2*)(xrow + k0);
        a.x = av.x; a.y = av.y;

        const float* w0 = W + (size_t)(k0 + 2 * hi) * N_HIDDEN;  // K = k0+2*hi
        const float* w1 = w0 + N_HIDDEN;                          // K = k0+2*hi+1

        v2f b0, b1, b2, b3;
        b0.x = w0[lo];      b0.y = w1[lo];
        b1.x = w0[16 + lo]; b1.y = w1[16 + lo];
        b2.x = w0[32 + lo]; b2.y = w1[32 + lo];
        b3.x = w0[48 + lo]; b3.y = w1[48 + lo];

        acc0 = __builtin_amdgcn_wmma_f32_16x16x4_f32(false, a, false, b0, (short)0, acc0, false, false);
        acc1 = __builtin_amdgcn_wmma_f32_16x16x4_f32(false, a, false, b1, (short)0, acc1, false, false);
        acc2 = __builtin_amdgcn_wmma_f32_16x16x4_f32(false, a, false, b2, (short)0, acc2, false, false);
        acc3 = __builtin_amdgcn_wmma_f32_16x16x4_f32(false, a, false, b3, (short)0, acc3, false, false);
    }

#pragma unroll
    for (int r = 0; r < 8; ++r) {
        const int m = row0 + r + 8 * hi;
        float* o = Wh + (size_t)m * N_HIDDEN + lo;
        o[0]  = acc0[r];
        o[16] = acc1[r];
        o[32] = acc2[r];
        o[48] = acc3[r];
    }
}

// ---------------------------------------------------------------------------
// K2: per-node partial attention dots: s1[n]=Wh[n]·a_w[:64], s2[n]=Wh[n]·a_w[64:]
// One wave per node; lane handles 2 contiguous elements; shuffle reduction.
// ---------------------------------------------------------------------------
__global__ __launch_bounds__(256) void node_scores(const float* __restrict__ Wh,
                                                   const float* __restrict__ aw,
                                                   float* __restrict__ s1,
                                                   float* __restrict__ s2) {
    const int wave = threadIdx.x >> 5;
    const int lane = threadIdx.x & 31;
    const int n = blockIdx.x * 8 + wave;
    if (n >= N_NODES) return;

    const float2 w  = *(const float2*)(Wh + (size_t)n * N_HIDDEN + 2 * lane);
    const float2 a1 = *(const float2*)(aw + 2 * lane);
    const float2 a2 = *(const float2*)(aw + N_HIDDEN + 2 * lane);

    float p1 = w.x * a1.x + w.y * a1.y;
    float p2 = w.x * a2.x + w.y * a2.y;

#pragma unroll
    for (int off = 16; off > 0; off >>= 1) {
        p1 += __shfl_down(p1, off, 32);
        p2 += __shfl_down(p2, off, 32);
    }
    if (lane == 0) { s1[n] = p1; s2[n] = p2; }
}

// ---------------------------------------------------------------------------
// K3: per-edge score -> h = exp(leaky_relu(s1[src]+s2[dst]+b)); scatter h_sum[src]
// ---------------------------------------------------------------------------
__global__ __launch_bounds__(256) void edge_score(const long long* __restrict__ ei,
                                                  const float* __restrict__ s1,
                                                  const float* __restrict__ s2,
                                                  const float* __restrict__ ab,
                                                  float* __restrict__ h_edge,
                                                  float* __restrict__ h_sum) {
    const int e = blockIdx.x * blockDim.x + threadIdx.x;
    if (e >= N_EDGES) return;
    const int src = (int)ei[e];
    const int dst = (int)ei[N_EDGES + e];
    const float score = s1[src] + s2[dst] + ab[0];
    const float lr = score > 0.0f ? score : NEG_SLOPE * score;
    const float h = __expf(lr);
    h_edge[e] = h;
    unsafeAtomicAdd(&h_sum[src], h);
}

// ---------------------------------------------------------------------------
// K4: per-edge aggregation: alpha = h/h_sum[src]; out[src] += alpha * Wh[dst]
// One wave per edge: lane handles 2 contiguous hidden components (coalesced
// 256B gather of Wh[dst], 64 hardware f32 atomics into out[src]).
// ---------------------------------------------------------------------------
__global__ __launch_bounds__(256) void edge_aggregate(const long long* __restrict__ ei,
                                                      const float* __restrict__ Wh,
                                                      const float* __restrict__ h_edge,
                                                      const float* __restrict__ h_sum,
                                                      float* __restrict__ out,
                                                      float* __restrict__ alpha) {
    const int wave = threadIdx.x >> 5;
    const int lane = threadIdx.x & 31;
    const long long e = (long long)blockIdx.x * 8 + wave;
    if (e >= N_EDGES) return;

    const int src = (int)ei[e];
    const int dst = (int)ei[N_EDGES + e];

    const float a = h_edge[e] / h_sum[src];
    if (lane == 0) alpha[e] = a;

    const float2 w = *(const float2*)(Wh + (size_t)dst * N_HIDDEN + 2 * lane);
    float* o = out + (size_t)src * N_HIDDEN + 2 * lane;
    unsafeAtomicAdd(o,     a * w.x);
    unsafeAtomicAdd(o + 1, a * w.y);
}

// ---------------------------------------------------------------------------
extern "C" void kernel_launch(void* const* d_in, const int* in_sizes, int n_in,
                              void* d_out, int out_size, void* d_ws, size_t ws_size,
                              hipStream_t stream) {
    const float*     x  = (const float*)d_in[0];       // [N, 256]
    const float*     W  = (const float*)d_in[1];       // [256, 64]
    const float*     aw = (const float*)d_in[2];       // [128]
    const float*     ab = (const float*)d_in[3];       // [1]
    const long long* ei = (const long long*)d_in[4];   // [2, E] int64

    float* out   = (float*)d_out;                      // [N, 64]
    float* alpha = out + (size_t)N_NODES * N_HIDDEN;   // [E]

    char*  ws     = (char*)d_ws;
    float* Wh     = (float*)ws;                                          // N*64
    float* s1     = (float*)(ws + (size_t)N_NODES * N_HIDDEN * sizeof(float));
    float* s2     = s1 + N_NODES;
    float* h_sum  = s2 + N_NODES;
    float* h_edge = h_sum + N_NODES;                                     // E

    hipMemsetAsync(out,   0, (size_t)N_NODES * N_HIDDEN * sizeof(float), stream);
    hipMemsetAsync(h_sum, 0, (size_t)N_NODES * sizeof(float), stream);

    const int gemm_blocks = ((N_NODES / 16) + 7) / 8;          // 8 waves/block, 1 tile/wave
    gemm_wh<<<gemm_blocks, 256, 0, stream>>>(x, W, Wh);

    node_scores<<<(N_NODES + 7) / 8, 256, 0, stream>>>(Wh, aw, s1, s2);

    edge_score<<<(N_EDGES + 255) / 256, 256, 0, stream>>>(ei, s1, s2, ab, h_edge, h_sum);

    edge_aggregate<<<(N_EDGES + 7) / 8, 256, 0, stream>>>(ei, Wh, h_edge, h_sum, out, alpha);
}